// SharedFusionNetwork_90752658964785
// MI455X (gfx1250) — compile-verified
//
#include <hip/hip_runtime.h>
#include <hip/hip_bf16.h>

// B=4096, M=64, D=512, DOUT=256, NH=8, HD=64
#define BB 4096
#define MM 64
#define DD 512
#define DOUT 256
#define NH 8
#define HD 64

typedef __attribute__((ext_vector_type(2))) float v2f;
typedef __attribute__((ext_vector_type(8))) float v8f;
typedef __attribute__((ext_vector_type(4))) unsigned int u32x4;
typedef __attribute__((ext_vector_type(8))) int i32x8;
typedef __attribute__((ext_vector_type(4))) int i32x4;

#if defined(__has_builtin)
#if __has_builtin(__builtin_amdgcn_tensor_load_to_lds)
#define USE_TDM 1
#endif
#endif
#ifndef USE_TDM
#define USE_TDM 0
#endif

// ---------------------------------------------------------------------------
// kP1: q = agg @ Wq^T + bq ;  qk[h,d] = (1/8) * sum_j q[h*64+j]*Wk[h*64+j][d]
//      sb[h] = (1/8) * q_h . bk_h
// ---------------------------------------------------------------------------
__global__ __launch_bounds__(512) void kP1(const float* __restrict__ agg,
                                           const float* __restrict__ ipw,
                                           const float* __restrict__ ipb,
                                           float* __restrict__ qk,
                                           float* __restrict__ sb) {
  __shared__ float qv[512];
  const int t = threadIdx.x;
  {
    float acc = ipb[t];
    const float* wr = ipw + (size_t)t * DD;
    for (int i = 0; i < DD; ++i) acc += agg[i] * wr[i];
    qv[t] = acc;
  }
  __syncthreads();
  for (int h = 0; h < NH; ++h) {
    float acc = 0.0f;
    const float* wk = ipw + (size_t)(DD + h * HD) * DD + t;  // column t, coalesced
    const float* qh = qv + h * HD;
    for (int j = 0; j < HD; ++j) acc += qh[j] * wk[(size_t)j * DD];
    qk[h * DD + t] = 0.125f * acc;  // fold 1/sqrt(HD)
  }
  if (t < NH) {
    float acc = 0.0f;
    for (int j = 0; j < HD; ++j) acc += qv[t * HD + j] * ipb[DD + t * HD + j];
    sb[t] = 0.125f * acc;
  }
}

// ---------------------------------------------------------------------------
// kP2: transpose weights so GEMM B-operands are [K][N] row-major (coalesced N)
// ---------------------------------------------------------------------------
__global__ void kP2(const float* __restrict__ ipw, const float* __restrict__ outw,
                    const float* __restrict__ w1, const float* __restrict__ w2,
                    float* __restrict__ WvT, float* __restrict__ WoutT,
                    float* __restrict__ W1T, float* __restrict__ W2T) {
  const int N1 = DD * DD;
  const int total = 3 * N1 + DOUT * DD;
  for (int idx = blockIdx.x * blockDim.x + threadIdx.x; idx < total;
       idx += gridDim.x * blockDim.x) {
    if (idx < N1) {
      int d = idx >> 9, o = idx & 511;
      WvT[idx] = ipw[(size_t)(2 * DD + o) * DD + d];
    } else if (idx < 2 * N1) {
      int k = idx - N1; int d = k >> 9, o = k & 511;
      WoutT[k] = outw[(size_t)o * DD + d];
    } else if (idx < 3 * N1) {
      int k = idx - 2 * N1; int d = k >> 9, o = k & 511;
      W1T[k] = w1[(size_t)o * DD + d];
    } else {
      int k = idx - 3 * N1; int d = k >> 8, o = k & 255;  // W2T[d*256+o]
      W2T[k] = w2[(size_t)o * DD + d];
    }
  }
}

// ---------------------------------------------------------------------------
// kA: per-batch fused  (stream H[b] exactly once from HBM into LDS via TDM)
//   scores -> softmax -> hbar & z_weighted -> ctx = hbar @ Wv^T + bv
//
// LDS layout of H tile: TDM pads 4 dwords every 256 dwords, i.e. element
// (m,d) lives at  m*520 + d + 4*(d>>8)  floats.  Row-to-row bank step = 8,
// mid-row pad keeps d-major segments conflict-free too.
// ---------------------------------------------------------------------------
#define HS 520

__global__ __launch_bounds__(256) void kA(const float* __restrict__ H,
                                          const float* __restrict__ w,
                                          const float* __restrict__ qk,
                                          const float* __restrict__ sb,
                                          const float* __restrict__ WvT,
                                          const float* __restrict__ ipb,
                                          float* __restrict__ ctx,
                                          float* __restrict__ zw) {
  __shared__ float sm[MM * HS + 4096 + 9 * MM + MM];  // ~152 KB (WGP LDS: 320 KB)
  float* Hs   = sm;                    // [64][520] padded TDM layout, LDS offset 0
  float* qks  = sm + MM * HS;          // [8][512], reused as hbar after softmax
  float* attn = qks + 4096;            // [9][64]  (head 8 = raw weights w)
  float* wls  = attn + 9 * MM;         // [64]
  const int t = threadIdx.x;
  const int b = blockIdx.x;
  const float* Hb = H + (size_t)b * MM * DD;

#if USE_TDM
  // -------- Tensor Data Mover: DMA 64x512 f32 tile -> LDS (padded) --------
  if (t < 32) {
    const unsigned long long ga = (unsigned long long)(const void*)Hb;
    u32x4 g0;
    g0.x = 1u;                                   // count=1, no gather
    g0.y = 0u;                                   // lds_addr = 0 (Hs is first LDS obj)
    g0.z = (unsigned int)(ga & 0xFFFFFFFFull);   // global_addr[31:0]
    g0.w = (unsigned int)((ga >> 32) & 0x1FFFFFFull) | 0x80000000u;  // type=2
    i32x8 g1;
    g1[0] = (2 << 16)        // data_size = 4B
          | (1 << 20)        // pad_enable
          | (7 << 22)        // pad_interval: 256 dwords
          | (3 << 25);       // pad_amount: 4 dwords
    g1[1] = (DD & 0xFFFF) << 16;   // tensor_dim0 = 512 into bits [63:48] of group1
    g1[2] = (MM << 16);            // tensor_dim0 hi16=0 | tensor_dim1 = 64
    g1[3] = (DD << 16);            // tensor_dim1 hi16=0 | tile_dim0 = 512
    g1[4] = MM;                    // tile_dim1 = 64 | tile_dim2 = 0
    g1[5] = DD;                    // tensor_dim0_stride = 512
    g1[6] = 0;                     // stride hi | tensor_dim1_stride lo
    g1[7] = 0;
    i32x4 z4 = {0, 0, 0, 0};
    i32x8 z8 = {0, 0, 0, 0, 0, 0, 0, 0};
    __builtin_amdgcn_tensor_load_to_lds(g0, g1, z4, z4, z8, 0);
#if __has_builtin(__builtin_amdgcn_s_wait_tensorcnt)
    __builtin_amdgcn_s_wait_tensorcnt(0);
#else
    asm volatile("s_wait_tensorcnt 0x0" ::: "memory");
#endif
  }
#else
  // -------- fallback: float4 loads into the same padded layout --------
  for (int i = 0; i < 32; ++i) {
    int idx = t + 256 * i;
    int m = idx >> 7;
    int c = (idx & 127) << 2;
    float4 v = *(const float4*)(Hb + (size_t)m * DD + c);
    float* dst = Hs + m * HS + c + ((c >> 8) << 2);
    dst[0] = v.x; dst[1] = v.y; dst[2] = v.z; dst[3] = v.w;
  }
#endif
  for (int i = 0; i < 16; ++i) qks[t + 256 * i] = qk[t + 256 * i];
  if (t < MM) wls[t] = w[b * MM + t];
  __syncthreads();

  // scores[h][m] = sb[h] + H[m,:] . qk[h,:]   (t -> m=t>>2, h = (t&3), (t&3)+4)
  {
    const int m = t >> 2, hh = t & 3;
    const float* hr = Hs + m * HS;
    for (int p = 0; p < 2; ++p) {
      const int h = hh + 4 * p;
      const float* qr = qks + h * DD;
      float acc = sb[h];
      for (int d = 0; d < 256; ++d) acc += hr[d] * qr[d];
      for (int d = 256; d < 512; ++d) acc += hr[d + 4] * qr[d];
      attn[h * MM + m] = acc;
    }
  }
  __syncthreads();

  // softmax per head (8 lanes), head 8 gets raw weights w
  if (t < NH) {
    float* a = attn + t * MM;
    float mx = a[0];
    for (int m = 1; m < MM; ++m) mx = fmaxf(mx, a[m]);
    float s = 0.0f;
    for (int m = 0; m < MM; ++m) { float e = __expf(a[m] - mx); a[m] = e; s += e; }
    float inv = 1.0f / s;
    for (int m = 0; m < MM; ++m) a[m] *= inv;
  } else if (t >= NH && t < NH + MM) {
    attn[NH * MM + (t - NH)] = wls[t - NH];
  }
  __syncthreads();

  // hbar[h][d] = sum_m attn[h][m]*H[m][d]  (heads 0..7 -> LDS; head 8 -> zw)
  float* hbar = qks;  // overlay: qk dead after scores barrier
  for (int h = 0; h < 9; ++h) {
    const float* a = attn + h * MM;
    for (int p = 0; p < 2; ++p) {
      const int d = t + 256 * p;
      const int dd = d + 4 * p;       // padded offset (p=1 <=> d>=256)
      float acc = 0.0f;
      for (int m = 0; m < MM; ++m) acc += a[m] * Hs[m * HS + dd];
      if (h < NH) hbar[h * DD + d] = acc;
      else        zw[(size_t)b * DD + d] = acc;
    }
  }
  __syncthreads();

  // ctx[o] = hbar[o>>6,:] . Wv[o,:] + bv[o]   (WvT coalesced over o)
  for (int p = 0; p < 2; ++p) {
    const int o = t + 256 * p;
    const float* hb = hbar + (o >> 6) * DD;
    const float* wv = WvT + o;
    float acc = ipb[2 * DD + o];
    for (int d = 0; d < DD; ++d) acc += hb[d] * wv[(size_t)d * DD];
    ctx[(size_t)b * DD + o] = acc;
  }
}

// ---------------------------------------------------------------------------
// WMMA f32 16x16x4 tile GEMM: A from LDS (16 rows, stride as), B = BT[k][n]
// ---------------------------------------------------------------------------
__device__ inline v8f wmma_tile(const float* Asm, int as,
                                const float* __restrict__ BT, int ntot,
                                int nbase, int K) {
  const int lane = threadIdx.x & 31;
  const int half = lane >> 4;
  const int r = lane & 15;
  v8f acc = {};
  for (int k = 0; k < K; k += 4) {
    v2f a, bfr;
    const float* ap = Asm + r * as + k + 2 * half;
    a.x = ap[0]; a.y = ap[1];
    const float* bp = BT + (size_t)(k + 2 * half) * ntot + nbase + r;
    bfr.x = bp[0]; bfr.y = bp[ntot];
    acc = __builtin_amdgcn_wmma_f32_16x16x4_f32(false, a, false, bfr,
                                                (short)0, acc, false, false);
  }
  return acc;
}

// ---------------------------------------------------------------------------
// kB: 32 batch rows per block.
//   z = ctx@WoutT + zw + outb ; LayerNorm ; gelu(z@W1T+b1) ; @W2T+b2 -> out
// ---------------------------------------------------------------------------
#define RB 32
#define BS 520

__global__ __launch_bounds__(256) void kB(const float* __restrict__ ctx,
                                          const float* __restrict__ zw,
                                          const float* __restrict__ WoutT,
                                          const float* __restrict__ outb,
                                          const float* __restrict__ lng,
                                          const float* __restrict__ lnb,
                                          const float* __restrict__ W1T,
                                          const float* __restrict__ b1,
                                          const float* __restrict__ W2T,
                                          const float* __restrict__ b2,
                                          float* __restrict__ out) {
  __shared__ float sm[2 * RB * BS + 2 * 256 + 64];  // ~135 KB
  float* cs   = sm;               // A buffer: ctx, later zn
  float* zb   = sm + RB * BS;     // z, later a1
  float* psum = sm + 2 * RB * BS; // 256
  float* psq  = psum + 256;       // 256
  float* mu   = psq + 256;        // 32
  float* rst  = mu + 32;          // 32
  const int t = threadIdx.x;
  const int b0 = blockIdx.x * RB;
  const int wv = t >> 5, lane = t & 31, half = lane >> 4, r = lane & 15;
  const int mt = wv & 1;    // which 16-row half of the 32-row tile
  const int ng = wv >> 1;   // 4 wave-groups over N

  // stage ctx tile (32x512)
  for (int i = 0; i < 16; ++i) {
    int idx = t + 256 * i;
    int row = idx >> 7;
    int c = (idx & 127) << 2;
    float4 v = *(const float4*)(ctx + (size_t)(b0 + row) * DD + c);
    float* dst = cs + row * BS + c;
    dst[0] = v.x; dst[1] = v.y; dst[2] = v.z; dst[3] = v.w;
  }
  __syncthreads();

  // GEMM1: z = ctx @ WoutT + zw + outb
  for (int i = 0; i < 8; ++i) {
    const int nb = (ng + 4 * i) * 16;
    v8f acc = wmma_tile(cs + mt * 16 * BS, BS, WoutT, DD, nb, DD);
    for (int vr = 0; vr < 8; ++vr) {
      const int M = mt * 16 + vr + 8 * half;
      const int N = nb + r;
      zb[M * BS + N] = acc[vr] + zw[(size_t)(b0 + M) * DD + N] + outb[N];
    }
  }
  __syncthreads();

  // LayerNorm: partials (8 threads per row x 64 elems), then finalize
  {
    const int row = t >> 3, seg = t & 7;
    const float* zr = zb + row * BS + seg * 64;
    float s = 0.0f, sq = 0.0f;
    for (int i = 0; i < 64; ++i) { float x = zr[i]; s += x; sq += x * x; }
    psum[t] = s; psq[t] = sq;
  }
  __syncthreads();
  if (t < RB) {
    float s = 0.0f, sq = 0.0f;
    for (int i = 0; i < 8; ++i) { s += psum[t * 8 + i]; sq += psq[t * 8 + i]; }
    const float m = s * (1.0f / 512.0f);
    const float v = sq * (1.0f / 512.0f) - m * m;
    mu[t] = m; rst[t] = rsqrtf(v + 1e-5f);
  }
  __syncthreads();
  {
    const int row = t >> 3, seg = t & 7;
    const float m = mu[row], rs = rst[row];
    const float* zr = zb + row * BS + seg * 64;
    float* cr = cs + row * BS + seg * 64;
    const float* g = lng + seg * 64;
    const float* be = lnb + seg * 64;
    for (int i = 0; i < 64; ++i) cr[i] = (zr[i] - m) * rs * g[i] + be[i];
  }
  __syncthreads();

  // GEMM2: a1 = gelu_exact(zn @ W1T + b1)
  for (int i = 0; i < 8; ++i) {
    const int nb = (ng + 4 * i) * 16;
    v8f acc = wmma_tile(cs + mt * 16 * BS, BS, W1T, DD, nb, DD);
    for (int vr = 0; vr < 8; ++vr) {
      const int M = mt * 16 + vr + 8 * half;
      const int N = nb + r;
      const float x = acc[vr] + b1[N];
      zb[M * BS + N] = 0.5f * x * (1.0f + erff(x * 0.70710678118654752f));
    }
  }
  __syncthreads();

  // GEMM3: out = a1 @ W2T + b2, nan_to_num
  for (int i = 0; i < 4; ++i) {
    const int nb = (ng + 4 * i) * 16;
    v8f acc = wmma_tile(zb + mt * 16 * BS, BS, W2T, DOUT, nb, DD);
    for (int vr = 0; vr < 8; ++vr) {
      const int M = mt * 16 + vr + 8 * half;
      const int N = nb + r;
      float x = acc[vr] + b2[N];
      x = (x != x) ? 0.0f : x;
      x = fminf(fmaxf(x, -3.3e38f), 3.3e38f);
      out[(size_t)(b0 + M) * DOUT + N] = x;
    }
  }
}

// ---------------------------------------------------------------------------
extern "C" void kernel_launch(void* const* d_in, const int* in_sizes, int n_in,
                              void* d_out, int out_size, void* d_ws, size_t ws_size,
                              hipStream_t stream) {
  const float* H    = (const float*)d_in[0];
  const float* w    = (const float*)d_in[1];
  const float* agg  = (const float*)d_in[2];
  const float* ipw  = (const float*)d_in[3];
  const float* ipb  = (const float*)d_in[4];
  const float* outw = (const float*)d_in[5];
  const float* outb = (const float*)d_in[6];
  const float* lng  = (const float*)d_in[7];
  const float* lnb  = (const float*)d_in[8];
  const float* w1   = (const float*)d_in[9];
  const float* b1   = (const float*)d_in[10];
  const float* w2   = (const float*)d_in[11];
  const float* b2   = (const float*)d_in[12];

  float* ws    = (float*)d_ws;
  float* qk    = ws;                    // 4096
  float* sb    = qk + 4096;             // 8 (pad 64)
  float* WvT   = sb + 64;               // 262144
  float* WoutT = WvT + 262144;          // 262144
  float* W1T   = WoutT + 262144;        // 262144
  float* W2T   = W1T + 262144;          // 131072
  float* ctx   = W2T + 131072;          // 4096*512
  float* zw    = ctx + (size_t)BB * DD; // 4096*512   total ~20.5 MB

  kP1<<<1, 512, 0, stream>>>(agg, ipw, ipb, qk, sb);
  kP2<<<256, 256, 0, stream>>>(ipw, outw, w1, w2, WvT, WoutT, W1T, W2T);
  kA<<<BB, 256, 0, stream>>>(H, w, qk, sb, WvT, ipb, ctx, zw);
  kB<<<BB / RB, 256, 0, stream>>>(ctx, zw, WoutT, outb, lng, lnb, W1T, b1, W2T, b2,
                                  (float*)d_out);
}